// Pipeline_29360396435747
// MI455X (gfx1250) — compile-verified
//
#include <hip/hip_runtime.h>
#include <hip/hip_bf16.h>
#include <math.h>

// ---------------------------------------------------------------------------
// MI455X / gfx1250 implementation. fp32 matrix path via V_WMMA_F32_16X16X4_F32.
// Double-buffered LDS staging (1 barrier / K-step), clamped weight loads
// (no divergent EXEC in the hot loop).
// ---------------------------------------------------------------------------

typedef __attribute__((ext_vector_type(2))) float v2f;
typedef __attribute__((ext_vector_type(4))) float v4f;
typedef __attribute__((ext_vector_type(8))) float v8f;

#define BATCH 8
#define SMAXC 4096
#define DFEAT 528

// GEMM tiling: 256 threads = 8 waves (4 M-waves x 2 N-waves), block tile 64x64.
#define BM 64
#define BN 64
#define BK 16
#define LSTR 20   // LDS row stride in floats (16B-aligned v4 stores, conflict-free)

// ---------------------------------------------------------------------------
// Utility kernels
// ---------------------------------------------------------------------------
__global__ void k_zero(float* __restrict__ p, size_t n) {
  size_t i = (size_t)blockIdx.x * blockDim.x + threadIdx.x;
  size_t stride = (size_t)gridDim.x * blockDim.x;
  for (; i < n; i += stride) p[i] = 0.0f;
}

// Scatter geo rows (528 f32) + pcd rows (3 f32) into dense padded tensors.
// Writes unscaled copy to ws (for GEMMs) and scaled copy to d_out.
__global__ void k_scatter(const float* __restrict__ geo, const float* __restrict__ pcd,
                          const int* __restrict__ indc, const int* __restrict__ inds,
                          int nrows, float* __restrict__ wsf, float* __restrict__ outf,
                          float* __restrict__ outp, float inv) {
  int i = blockIdx.x;
  if (i >= nrows) return;
  int dst = inds[i];
  int src = indc[i];
  const float* g = geo + (size_t)src * DFEAT;
  float* w = wsf + (size_t)dst * DFEAT;
  float* o = outf + (size_t)dst * DFEAT;
  for (int j = threadIdx.x; j < DFEAT; j += blockDim.x) {
    float v = g[j];
    w[j] = v;
    o[j] = v * inv;
  }
  if (threadIdx.x < 3) outp[(size_t)dst * 3 + threadIdx.x] = pcd[(size_t)src * 3 + threadIdx.x];
}

// Per-batch mask counts (float): lens[0..7]=src, lens[8..15]=tgt.
__global__ void k_lens(const unsigned char* __restrict__ sm,
                       const unsigned char* __restrict__ tm,
                       float* __restrict__ lens) {
  __shared__ float sh[256];
  int b = blockIdx.x;  // 0..15
  const unsigned char* m = (b < 8) ? (sm + (size_t)b * SMAXC) : (tm + (size_t)(b - 8) * SMAXC);
  float s = 0.0f;
  for (int j = threadIdx.x; j < SMAXC; j += 256) s += (float)m[j];
  sh[threadIdx.x] = s;
  __syncthreads();
  for (int o = 128; o > 0; o >>= 1) {
    if (threadIdx.x < o) sh[threadIdx.x] += sh[threadIdx.x + o];
    __syncthreads();
  }
  if (threadIdx.x == 0) lens[b] = sh[0];
}

// glob[b][o] = bs1[o] + (inst_acc[b]/len_s[b]) . Ws1[o,528:1552]
//                     + (cat_acc[b]/len_t[b]) . Ws1[o,1552:2576]
__global__ void k_glob(const float* __restrict__ Ws1, const float* __restrict__ bs1,
                       const float* __restrict__ iacc, const float* __restrict__ cacc,
                       const float* __restrict__ lens, float* __restrict__ glob) {
  int idx = blockIdx.x * blockDim.x + threadIdx.x;  // 8*512
  int b = idx >> 9;
  int o = idx & 511;
  const float* wi = Ws1 + (size_t)o * 2576 + DFEAT;
  const float* wc = Ws1 + (size_t)o * 2576 + DFEAT + 1024;
  const float* ia = iacc + (size_t)b * 1024;
  const float* ca = cacc + (size_t)b * 1024;
  float si = 0.0f, sc = 0.0f;
  for (int k = 0; k < 1024; ++k) {
    si += ia[k] * wi[k];
    sc += ca[k] * wc[k];
  }
  glob[idx] = si / lens[b] + sc / lens[8 + b] + bs1[o];
}

// scale_mat: one wave per row, dot(z2_row[256], Ws3) + bs3, sigmoid-0.5.
__global__ __launch_bounds__(256)
void k_score(const float* __restrict__ z2, const float* __restrict__ Ws3,
             const float* __restrict__ bs3, float* __restrict__ outs) {
  int wave = threadIdx.x >> 5;
  int lane = threadIdx.x & 31;
  int row = blockIdx.x * 8 + wave;  // grid 4096 -> 32768 rows
  const v4f* zr = (const v4f*)(z2 + (size_t)row * 256);
  const v4f* wr = (const v4f*)Ws3;
  v4f za = zr[lane * 2], zb = zr[lane * 2 + 1];
  v4f wa = wr[lane * 2], wb = wr[lane * 2 + 1];
  float s = za.x * wa.x + za.y * wa.y + za.z * wa.z + za.w * wa.w +
            zb.x * wb.x + zb.y * wb.y + zb.z * wb.z + zb.w * wb.w;
  for (int off = 16; off > 0; off >>= 1) s += __shfl_xor(s, off, 32);
  if (lane == 0) {
    float v = s + bs3[0];
    outs[row] = 1.0f / (1.0f + __expf(-v)) - 0.5f;
  }
}

// ---------------------------------------------------------------------------
// fp32 WMMA GEMM:  C[M,N] = relu(A[M,K] * W[N,ldw]^T + bias[n] (+ perb[b,n]))
// If racc != null: instead of storing C, masked-sum rows into racc[b*N+n]
// (mask is one byte per global row). M multiple of 64, K multiple of 16, N>=1.
// Out-of-range weight rows are clamped (their products land only in columns
// n>=N, which the epilogue never reads/stores).
// ---------------------------------------------------------------------------
__global__ __launch_bounds__(256)
void k_gemm(const float* __restrict__ A, const float* __restrict__ W, int ldw,
            const float* __restrict__ bias, const float* __restrict__ perb,
            float* __restrict__ C, float* __restrict__ racc,
            const unsigned char* __restrict__ mask,
            int M, int N, int K) {
  __shared__ float As[2][BM * LSTR];
  __shared__ float Bs[2][BN * LSTR];
  __shared__ float s_maskf[BM];
  __shared__ float s_red[BN];

  const int t = threadIdx.x;
  const int lane = t & 31;
  const int wave = t >> 5;
  const int wm = (wave & 3) * 16;   // wave M offset in tile
  const int wn = (wave >> 2) * 32;  // wave N offset in tile (2x 16-wide subtiles)
  const int n0 = blockIdx.x * BN;
  const int m0 = blockIdx.y * BM;

  if (racc) {
    if (t < BM) s_maskf[t] = (float)mask[m0 + t];
    if (t < BN) s_red[t] = 0.0f;
  }

  v8f acc0 = {};
  v8f acc1 = {};

  const int lrow = t >> 2;           // 0..63 (tile row for staging)
  const int kc = (t & 3) * 4;        // 0,4,8,12
  const int l15 = lane & 15;
  const int hsel = (lane >> 4) * 2;  // 0 or 2 (K half-select per ISA A/B layout)

  const float* __restrict__ Arow = A + (size_t)(m0 + lrow) * K;
  const int brow = (n0 + lrow < N) ? (n0 + lrow) : (N - 1);  // clamp: no divergence
  const float* __restrict__ Wrow = W + (size_t)brow * ldw;

  const int nt = K / BK;

  // Prologue: stage tile 0 into buffer 0.
  {
    v4f av = *(const v4f*)(Arow + kc);
    v4f bv = *(const v4f*)(Wrow + kc);
    *(v4f*)&As[0][lrow * LSTR + kc] = av;
    *(v4f*)&Bs[0][lrow * LSTR + kc] = bv;
  }
  __syncthreads();

  for (int i = 0; i < nt; ++i) {
    const int cur = i & 1;
    v4f av, bv;
    const bool more = (i + 1) < nt;
    if (more) {
      const int k0 = (i + 1) * BK;
      av = *(const v4f*)(Arow + k0 + kc);
      bv = *(const v4f*)(Wrow + k0 + kc);
    }

#pragma unroll
    for (int kk = 0; kk < 4; ++kk) {
      v2f af = *(const v2f*)&As[cur][(wm + l15) * LSTR + kk * 4 + hsel];
      v2f bf0 = *(const v2f*)&Bs[cur][(wn + l15) * LSTR + kk * 4 + hsel];
      v2f bf1 = *(const v2f*)&Bs[cur][(wn + 16 + l15) * LSTR + kk * 4 + hsel];
      acc0 = __builtin_amdgcn_wmma_f32_16x16x4_f32(false, af, false, bf0,
                                                   (short)0, acc0, false, false);
      acc1 = __builtin_amdgcn_wmma_f32_16x16x4_f32(false, af, false, bf1,
                                                   (short)0, acc1, false, false);
    }

    if (more) {
      *(v4f*)&As[cur ^ 1][lrow * LSTR + kc] = av;
      *(v4f*)&Bs[cur ^ 1][lrow * LSTR + kc] = bv;
    }
    __syncthreads();
  }

  const int mbase = wm + (lane >> 4) * 8;  // C/D layout: lanes 16-31 hold M+8
  if (!racc) {
#pragma unroll
    for (int v = 0; v < 8; ++v) {
      int mrow = m0 + mbase + v;
      int na = n0 + wn + l15;
      if (na < N) {
        float b = (bias ? bias[na] : 0.0f) +
                  (perb ? perb[(size_t)(mrow / SMAXC) * N + na] : 0.0f);
        C[(size_t)mrow * N + na] = fmaxf(acc0[v] + b, 0.0f);
      }
      int nb = n0 + wn + 16 + l15;
      if (nb < N) {
        float b = (bias ? bias[nb] : 0.0f) +
                  (perb ? perb[(size_t)(mrow / SMAXC) * N + nb] : 0.0f);
        C[(size_t)mrow * N + nb] = fmaxf(acc1[v] + b, 0.0f);
      }
    }
  } else {
    int na = n0 + wn + l15;
    int nb = n0 + wn + 16 + l15;
    float b0 = (na < N) ? bias[na] : 0.0f;
    float b1 = (nb < N) ? bias[nb] : 0.0f;
    float p0 = 0.0f, p1 = 0.0f;
#pragma unroll
    for (int v = 0; v < 8; ++v) {
      float mf = s_maskf[mbase + v];
      p0 += (na < N) ? fmaxf(acc0[v] + b0, 0.0f) * mf : 0.0f;
      p1 += (nb < N) ? fmaxf(acc1[v] + b1, 0.0f) * mf : 0.0f;
    }
    atomicAdd(&s_red[wn + l15], p0);
    atomicAdd(&s_red[wn + 16 + l15], p1);
    __syncthreads();
    if (t < BN && (n0 + t) < N) {
      int b = m0 / SMAXC;  // all 64 rows in this block share one batch (4096%64==0)
      atomicAdd(&racc[(size_t)b * N + n0 + t], s_red[t]);
    }
  }
}

// ---------------------------------------------------------------------------
// Host orchestration
// ---------------------------------------------------------------------------
extern "C" void kernel_launch(void* const* d_in, const int* in_sizes, int n_in,
                              void* d_out, int out_size, void* d_ws, size_t ws_size,
                              hipStream_t stream) {
  const float* geo = (const float*)d_in[0];
  const float* pcd = (const float*)d_in[1];
  const float* Wi1 = (const float*)d_in[2];
  const float* bi1 = (const float*)d_in[3];
  const float* Wi2 = (const float*)d_in[4];
  const float* bi2 = (const float*)d_in[5];
  const float* Wc1 = (const float*)d_in[6];
  const float* bc1 = (const float*)d_in[7];
  const float* Wc2 = (const float*)d_in[8];
  const float* bc2 = (const float*)d_in[9];
  const float* Ws1 = (const float*)d_in[10];
  const float* bs1 = (const float*)d_in[11];
  const float* Ws2 = (const float*)d_in[12];
  const float* bs2 = (const float*)d_in[13];
  const float* Ws3 = (const float*)d_in[14];
  const float* bs3 = (const float*)d_in[15];
  const unsigned char* smask = (const unsigned char*)d_in[16];
  const unsigned char* tmask = (const unsigned char*)d_in[17];
  const int* sics = (const int*)d_in[18];
  const int* tics = (const int*)d_in[19];
  const int* sic = (const int*)d_in[20];
  const int* tic = (const int*)d_in[21];
  const int Ns = in_sizes[20];
  const int Nt = in_sizes[21];

  const size_t FEAT = (size_t)BATCH * SMAXC * DFEAT;  // 17,301,504 floats
  const size_t PCD3 = (size_t)BATCH * SMAXC * 3;
  float* out = (float*)d_out;
  float* o_src = out;
  float* o_tgt = out + FEAT;
  float* o_spcd = out + 2 * FEAT;
  float* o_tpcd = o_spcd + PCD3;
  float* o_scale = o_tpcd + PCD3;

  // Workspace layout (~242 MB total)
  float* ws = (float*)d_ws;
  float* ws_src = ws;                              // [32768][528]
  float* ws_tgt = ws_src + FEAT;                   // [32768][528]
  float* ws_h = ws_tgt + FEAT;                     // reused: h1 / g1 / z1
  float* ws_z2 = ws_h + FEAT;                      // [32768][256]
  float* ws_iacc = ws_z2 + (size_t)32768 * 256;    // [8][1024]
  float* ws_cacc = ws_iacc + 8 * 1024;             // [8][1024]
  float* ws_glob = ws_cacc + 8 * 1024;             // [8][512]
  float* ws_lens = ws_glob + 8 * 512;              // [16]

  const float inv = 1.0f / sqrtf((float)DFEAT);
  const int M = BATCH * SMAXC;  // 32768

  // Zero-init dense outputs, scatter targets, reduction accumulators.
  k_zero<<<4096, 256, 0, stream>>>(out, 2 * FEAT + 2 * PCD3);
  k_zero<<<4096, 256, 0, stream>>>(ws_src, 2 * FEAT);
  k_zero<<<64, 256, 0, stream>>>(ws_iacc, (size_t)(8 * 1024 * 2 + 8 * 512 + 16));

  k_scatter<<<Ns, 256, 0, stream>>>(geo, pcd, sic, sics, Ns, ws_src, o_src, o_spcd, inv);
  k_scatter<<<Nt, 256, 0, stream>>>(geo, pcd, tic, tics, Nt, ws_tgt, o_tgt, o_tpcd, inv);
  k_lens<<<16, 256, 0, stream>>>(smask, tmask, ws_lens);

  // src MLP: h1 = relu(src @ Wi1^T + bi1); pooled h2 via fused masked reduce.
  k_gemm<<<dim3(9, 512), 256, 0, stream>>>(ws_src, Wi1, 528, bi1, nullptr,
                                           ws_h, nullptr, nullptr, M, 528, 528);
  k_gemm<<<dim3(16, 512), 256, 0, stream>>>(ws_h, Wi2, 528, bi2, nullptr,
                                            nullptr, ws_iacc, smask, M, 1024, 528);
  // tgt MLP (reuse ws_h for g1)
  k_gemm<<<dim3(9, 512), 256, 0, stream>>>(ws_tgt, Wc1, 528, bc1, nullptr,
                                           ws_h, nullptr, nullptr, M, 528, 528);
  k_gemm<<<dim3(16, 512), 256, 0, stream>>>(ws_h, Wc2, 528, bc2, nullptr,
                                            nullptr, ws_cacc, tmask, M, 1024, 528);

  k_glob<<<16, 256, 0, stream>>>(Ws1, bs1, ws_iacc, ws_cacc, ws_lens, ws_glob);

  // z1 = relu(src @ Ws1_s^T + glob[b])  (Ws1 row stride 2576, cols 0:528)
  k_gemm<<<dim3(8, 512), 256, 0, stream>>>(ws_src, Ws1, 2576, nullptr, ws_glob,
                                           ws_h, nullptr, nullptr, M, 512, 528);
  // z2 = relu(z1 @ Ws2^T + bs2)
  k_gemm<<<dim3(4, 512), 256, 0, stream>>>(ws_h, Ws2, 512, bs2, nullptr,
                                           ws_z2, nullptr, nullptr, M, 256, 512);
  // s -> sigmoid - 0.5 -> scale_mat
  k_score<<<4096, 256, 0, stream>>>(ws_z2, Ws3, bs3, o_scale);
}